// AttnDecoderRNN_56118042689791
// MI455X (gfx1250) — compile-verified
//
#include <hip/hip_runtime.h>
#include <hip/hip_bf16.h>

// AttnDecoderRNN single step, MI455X (gfx1250), wave32 + WMMA f32 16x16x4.
// Dominant cost: streaming out_W (206 MB) once -> ~8.8us at 23.3 TB/s. All
// GEMMs (comb, GRU ih/hh, output projection) use V_WMMA_F32_16X16X4_F32 with
// M=128 handled by 8 waves per workgroup sharing the B-operand via L0.

#define B_ 128
#define H_ 1024
#define L_ 60
#define V_ 50257

typedef __attribute__((ext_vector_type(2))) float v2f;
typedef __attribute__((ext_vector_type(8))) float v8f;

// ---------------------------------------------------------------------------
// K1: embedding gather + attention logits + softmax + context; emits
//     xcat[b] = [embedded(1024) | attn_applied(1024)] and attn_weights.
// One block per batch row, 256 threads (8 waves).
// ---------------------------------------------------------------------------
__global__ void k_embed_attn(const int* __restrict__ input,
                             const float* __restrict__ hidden,   // [B,H]
                             const float* __restrict__ enc,      // [B,L,H]
                             const float* __restrict__ emb,      // [V,H]
                             const float* __restrict__ attn_W,   // [L,2H]
                             const float* __restrict__ attn_b,   // [L]
                             float* __restrict__ xcat,           // [B,2H]
                             float* __restrict__ attn_w_out) {   // [B,L]
    __shared__ float s_cat[2 * H_];
    __shared__ float s_logit[64];
    __shared__ float s_w[64];

    const int b   = blockIdx.x;
    const int tid = threadIdx.x;
    const int tok = input[b];

    // stage embedded|h into LDS
    for (int j = tid; j < H_; j += 256) {
        s_cat[j]       = emb[(size_t)tok * H_ + j];
        s_cat[H_ + j]  = hidden[(size_t)b * H_ + j];
    }
    __syncthreads();

    // attention logits: wave w handles l = w, w+8, ... ; lane-strided dot(2048)
    const int lane = tid & 31;
    const int wave = tid >> 5;
    for (int l = wave; l < L_; l += 8) {
        const float* wrow = attn_W + (size_t)l * (2 * H_);
        float p = 0.f;
        for (int j = lane; j < 2 * H_; j += 32) p += s_cat[j] * wrow[j];
        for (int off = 16; off > 0; off >>= 1) p += __shfl_xor(p, off, 32);
        if (lane == 0) s_logit[l] = p + attn_b[l];
    }
    __syncthreads();

    // softmax over L=60 (redundant per-thread scan is cheap)
    if (tid < L_) {
        float mx = -3.4e38f;
        for (int l = 0; l < L_; ++l) mx = fmaxf(mx, s_logit[l]);
        float s = 0.f;
        for (int l = 0; l < L_; ++l) s += __expf(s_logit[l] - mx);
        float w = __expf(s_logit[tid] - mx) / s;
        s_w[tid] = w;
        attn_w_out[(size_t)b * L_ + tid] = w;
    }
    __syncthreads();

    // context vector + write xcat (coalesced over hh)
    for (int hh = tid; hh < H_; hh += 256) {
        const float* e = enc + (size_t)b * L_ * H_ + hh;
        float acc = 0.f;
        for (int l = 0; l < L_; ++l) acc += s_w[l] * e[(size_t)l * H_];
        xcat[(size_t)b * (2 * H_) + hh]      = s_cat[hh];
        xcat[(size_t)b * (2 * H_) + H_ + hh] = acc;
    }
}

// ---------------------------------------------------------------------------
// Generic fp32 WMMA GEMM: C[128,N] = A[128,K] * W[N,K]^T + bias (+relu).
// Block = 256 threads = 8 waves; wave w -> M rows [16w,16w+16); block -> one
// 16-wide N tile. K stepped by 4 via V_WMMA_F32_16X16X4_F32.
// Lane layout (ISA 7.12.2): A/B VGPR0/1, lanes 0-15 hold K=k,k+1 and lanes
// 16-31 hold K=k+2,k+3; C/D VGPR j -> row m0 + (lane>=16)*8 + j, col n0+lane%16.
// ---------------------------------------------------------------------------
__global__ void k_gemm_wmma(const float* __restrict__ A,
                            const float* __restrict__ W,
                            const float* __restrict__ bias,
                            float* __restrict__ C,
                            int N, int K, int relu) {
    const int tid  = threadIdx.x;
    const int lane = tid & 31;
    const int wave = tid >> 5;
    const int l15  = lane & 15;
    const int half = lane >> 4;          // 0: K+{0,1}, 1: K+{2,3}
    const int m0   = wave * 16;
    const int n0   = blockIdx.x * 16;
    const int nIdx = min(n0 + l15, N - 1);   // clamp tail loads, EXEC stays all-1s

    const float* aPtr = A + (size_t)(m0 + l15) * K + half * 2;
    const float* wPtr = W + (size_t)nIdx * K + half * 2;

    v8f acc = {};
#pragma unroll 8
    for (int k = 0; k < K; k += 4) {
        v2f a = *(const v2f*)(aPtr + k);   // A[m, k+2*half .. +1]
        v2f bb = *(const v2f*)(wPtr + k);  // W[n, k+2*half .. +1] == B[k..][n]
        acc = __builtin_amdgcn_wmma_f32_16x16x4_f32(
            /*neg_a=*/false, a, /*neg_b=*/false, bb,
            /*c_mod=*/(short)0, acc, /*reuse_a=*/false, /*reuse_b=*/false);
    }

    if (n0 + l15 < N) {
        const float bv = bias ? bias[n0 + l15] : 0.f;
#pragma unroll
        for (int j = 0; j < 8; ++j) {
            float v = acc[j] + bv;
            if (relu) v = fmaxf(v, 0.f);
            C[(size_t)(m0 + half * 8 + j) * N + (n0 + l15)] = v;
        }
    }
}

// ---------------------------------------------------------------------------
// K3b: GRU elementwise (PyTorch gate order r,z,n).
// ---------------------------------------------------------------------------
__global__ void k_gru(const float* __restrict__ gi,
                      const float* __restrict__ gh,
                      const float* __restrict__ hidden,
                      float* __restrict__ h_new) {
    const int i = blockIdx.x * blockDim.x + threadIdx.x;
    if (i >= B_ * H_) return;
    const int b = i / H_, j = i % H_;
    const float* gib = gi + (size_t)b * 3 * H_;
    const float* ghb = gh + (size_t)b * 3 * H_;
    const float ir = gib[j], iz = gib[H_ + j], in_ = gib[2 * H_ + j];
    const float hr = ghb[j], hz = ghb[H_ + j], hn  = ghb[2 * H_ + j];
    const float r = 1.f / (1.f + __expf(-(ir + hr)));
    const float z = 1.f / (1.f + __expf(-(iz + hz)));
    const float n = tanhf(in_ + r * hn);
    h_new[i] = (1.f - z) * n + z * hidden[i];
}

// ---------------------------------------------------------------------------
// K5: row-wise log_softmax over V=50257. One block per batch row.
// ---------------------------------------------------------------------------
__global__ void k_logsoftmax(const float* __restrict__ logits,
                             float* __restrict__ out) {
    __shared__ float red[256];
    const int b   = blockIdx.x;
    const int tid = threadIdx.x;
    const float* row = logits + (size_t)b * V_;

    float mx = -3.4e38f;
    for (int j = tid; j < V_; j += 256) mx = fmaxf(mx, row[j]);
    red[tid] = mx;
    __syncthreads();
    for (int s = 128; s > 0; s >>= 1) {
        if (tid < s) red[tid] = fmaxf(red[tid], red[tid + s]);
        __syncthreads();
    }
    mx = red[0];
    __syncthreads();

    float sm = 0.f;
    for (int j = tid; j < V_; j += 256) sm += __expf(row[j] - mx);
    red[tid] = sm;
    __syncthreads();
    for (int s = 128; s > 0; s >>= 1) {
        if (tid < s) red[tid] += red[tid + s];
        __syncthreads();
    }
    const float lse = mx + __logf(red[0]);

    float* orow = out + (size_t)b * V_;
    for (int j = tid; j < V_; j += 256) orow[j] = row[j] - lse;
}

// ---------------------------------------------------------------------------
extern "C" void kernel_launch(void* const* d_in, const int* in_sizes, int n_in,
                              void* d_out, int out_size, void* d_ws, size_t ws_size,
                              hipStream_t stream) {
    (void)in_sizes; (void)n_in; (void)out_size; (void)ws_size;

    const int*   input   = (const int*)  d_in[0];
    const float* hidden  = (const float*)d_in[1];   // [1,B,H] == [B,H]
    const float* enc     = (const float*)d_in[2];   // [B,L,H]
    const float* emb     = (const float*)d_in[3];   // [V,H]
    const float* attn_W  = (const float*)d_in[4];   // [L,2H]
    const float* attn_b  = (const float*)d_in[5];   // [L]
    const float* comb_W  = (const float*)d_in[6];   // [H,2H]
    const float* comb_b  = (const float*)d_in[7];   // [H]
    const float* W_ih    = (const float*)d_in[8];   // [3H,H]
    const float* W_hh    = (const float*)d_in[9];   // [3H,H]
    const float* b_ih    = (const float*)d_in[10];  // [3H]
    const float* b_hh    = (const float*)d_in[11];  // [3H]
    const float* out_W   = (const float*)d_in[12];  // [V,H]
    const float* out_b   = (const float*)d_in[13];  // [V]

    float* out    = (float*)d_out;
    float* logp   = out;                              // B*V
    float* h_new  = out + (size_t)B_ * V_;            // B*H (as [1,B,H])
    float* attn_w = h_new + (size_t)B_ * H_;          // B*L

    float* ws     = (float*)d_ws;
    float* xcat   = ws;                               // B*2H
    float* x      = xcat + (size_t)B_ * 2 * H_;       // B*H
    float* gi     = x    + (size_t)B_ * H_;           // B*3H
    float* gh     = gi   + (size_t)B_ * 3 * H_;       // B*3H
    float* logits = gh   + (size_t)B_ * 3 * H_;       // B*V

    // 1) embedding + attention + context
    k_embed_attn<<<B_, 256, 0, stream>>>(input, hidden, enc, emb, attn_W, attn_b,
                                         xcat, attn_w);
    // 2) x = relu(xcat @ comb_W^T + comb_b)     [128 x 1024], K=2048
    k_gemm_wmma<<<H_ / 16, 256, 0, stream>>>(xcat, comb_W, comb_b, x,
                                             H_, 2 * H_, 1);
    // 3) gi = x @ W_ih^T + b_ih                 [128 x 3072], K=1024
    k_gemm_wmma<<<(3 * H_) / 16, 256, 0, stream>>>(x, W_ih, b_ih, gi,
                                                   3 * H_, H_, 0);
    // 4) gh = h @ W_hh^T + b_hh                 [128 x 3072], K=1024
    k_gemm_wmma<<<(3 * H_) / 16, 256, 0, stream>>>(hidden, W_hh, b_hh, gh,
                                                   3 * H_, H_, 0);
    // 5) GRU elementwise -> h_new (straight into d_out)
    k_gru<<<(B_ * H_) / 256, 256, 0, stream>>>(gi, gh, hidden, h_new);
    // 6) logits = h_new @ out_W^T + out_b       [128 x 50257], K=1024 (206 MB stream)
    k_gemm_wmma<<<(V_ + 15) / 16, 256, 0, stream>>>(h_new, out_W, out_b, logits,
                                                    V_, H_, 0);
    // 7) log_softmax over vocab
    k_logsoftmax<<<B_, 256, 0, stream>>>(logits, logp);
}